// StyleLINKX_31774168056059
// MI455X (gfx1250) — compile-verified
//
#include <hip/hip_runtime.h>

#define N_NODES 65536
#define N_EDGES 1048576
#define IN_C    64
#define HID     128
#define OUT_C   64
#define EPSV    1e-5f
#define SLOPE   0.01f
#define LDS_STRIDE 132   // 132 % 64 == 4 -> conflict-free column walks

typedef __attribute__((ext_vector_type(2))) float v2f;
typedef __attribute__((ext_vector_type(8))) float v8f;

// ---------------------------------------------------------------- WMMA core
__device__ __forceinline__ v8f wmma4(v2f a, v2f b, v8f c) {
  // D = A(16x4 f32) x B(4x16 f32) + C(16x16 f32)
  return __builtin_amdgcn_wmma_f32_16x16x4_f32(
      /*neg_a=*/false, a, /*neg_b=*/false, b,
      /*c_mod=*/(short)0, c, /*reuse_a=*/false, /*reuse_b=*/false);
}

// 16xK tile (LDS, padded stride) times W[n0..n0+15][0..K-1]^T -> 16x16 acc.
__device__ __forceinline__ v8f gemm_tile(const float* __restrict__ A,
                                         const float* __restrict__ W,
                                         int K, int n0, int lane) {
  const int row = lane & 15;
  const int kk  = (lane >> 4) << 1;
  const float* __restrict__ ar = A + row * LDS_STRIDE + kk;
  const float* __restrict__ wr = W + (n0 + (lane & 15)) * K + kk;
  v8f acc = {};
#pragma unroll 4
  for (int k0 = 0; k0 < K; k0 += 4) {
    v2f a; a.x = ar[k0]; a.y = ar[k0 + 1];
    v2f b; b.x = wr[k0]; b.y = wr[k0 + 1];
    acc = wmma4(a, b, acc);
  }
  return acc;
}

// Per-row instance norm + style affine + leaky relu, in place on LDS tile.
__device__ __forceinline__ void norm_act(float* __restrict__ H, int C,
                                         const float* __restrict__ style,
                                         int lane) {
  const int r  = lane & 15;
  const int c0 = (lane >> 4) * (C >> 1);
  const int c1 = c0 + (C >> 1);
  float s1 = 0.f, s2 = 0.f;
  for (int c = c0; c < c1; ++c) {
    float v = H[r * LDS_STRIDE + c];
    s1 += v; s2 += v * v;
  }
  s1 += __shfl_xor(s1, 16, 32);
  s2 += __shfl_xor(s2, 16, 32);
  const float inv = 1.0f / (float)C;
  const float mu  = s1 * inv;
  const float var = s2 * inv - mu * mu;
  const float rs  = rsqrtf(var + EPSV);
  for (int c = c0; c < c1; ++c) {
    float v = (H[r * LDS_STRIDE + c] - mu) * rs;
    float y = style[c] * v + style[C + c];
    H[r * LDS_STRIDE + c] = fmaxf(y, SLOPE * y);   // leaky relu, slope<1
  }
}

// ------------------------------------------------- CSR segment-sum kernels
__global__ void __launch_bounds__(256)
zero_int(int* __restrict__ p) {
  p[blockIdx.x * 256 + threadIdx.x] = 0;
}

__global__ void __launch_bounds__(256)
count_deg(const int* __restrict__ ei, int* __restrict__ deg) {
  int e = blockIdx.x * 256 + threadIdx.x;
  atomicAdd(&deg[ei[N_EDGES + e]], 1);   // int atomics: native, 1 per edge
}

// Single-block exclusive scan of 65536 degrees (deg lives in `cursor`,
// overwritten in place with the running prefix; row_start gets the same).
__global__ void __launch_bounds__(1024)
scan_deg(int* __restrict__ cursor, int* __restrict__ row_start) {
  __shared__ int part[1024];
  const int t    = threadIdx.x;
  const int base = t * 64;
  int s = 0;
  for (int i = 0; i < 64; ++i) s += cursor[base + i];
  part[t] = s;
  __syncthreads();
  int pre = 0;
  for (int i = 0; i < t; ++i) pre += part[i];   // one-time, 1 block: fine
  int run = pre;
  for (int i = 0; i < 64; ++i) {
    int d = cursor[base + i];
    row_start[base + i] = run;
    cursor[base + i]    = run;
    run += d;
  }
  if (t == 1023) row_start[N_NODES] = run;      // == N_EDGES
}

__global__ void __launch_bounds__(256)
fill_csr(const int* __restrict__ ei, int* __restrict__ cursor,
         int* __restrict__ src_sorted) {
  int e = blockIdx.x * 256 + threadIdx.x;
  int s = ei[e];
  int d = ei[N_EDGES + e];
  int pos = atomicAdd(&cursor[d], 1);
  src_sorted[pos] = s;
}

// One wave per node: 32 lanes x float4 covers the 128-channel row.
// Inner loop: uniform index load + one 128-bit Wa row read (L2 resident).
__global__ void __launch_bounds__(256)
gather_sum(const int* __restrict__ row_start, const int* __restrict__ src_sorted,
           const float* __restrict__ Wa, const float* __restrict__ ba,
           float* __restrict__ agg) {
  const int n    = (blockIdx.x * 256 + threadIdx.x) >> 5;
  const int lane = threadIdx.x & 31;
  const int c4   = lane << 2;
  float4 acc = *(const float4*)&ba[c4];
  const int beg = row_start[n];
  const int end = row_start[n + 1];
  for (int j = beg; j < end; ++j) {
    const int s = src_sorted[j];
    const float4 v = *(const float4*)&Wa[(size_t)s * HID + c4];
    acc.x += v.x; acc.y += v.y; acc.z += v.z; acc.w += v.w;
  }
  *(float4*)&agg[(size_t)n * HID + c4] = acc;
}

// ---------------------------------------------------------------- styles
__global__ void __launch_bounds__(640)
style_kernel(const float* __restrict__ w,
             const float* __restrict__ aW0, const float* __restrict__ ab0,
             const float* __restrict__ aW1, const float* __restrict__ ab1,
             const float* __restrict__ aW2, const float* __restrict__ ab2,
             float* __restrict__ s0, float* __restrict__ s1,
             float* __restrict__ s2) {
  int t = threadIdx.x;
  const float* aW; const float* ab; float* s; int j;
  if (t < 256)      { aW = aW0; ab = ab0; s = s0; j = t;       }
  else if (t < 512) { aW = aW1; ab = ab1; s = s1; j = t - 256; }
  else              { aW = aW2; ab = ab2; s = s2; j = t - 512; }
  float acc = ab[j];
  for (int k = 0; k < 128; ++k) acc += w[k] * aW[j * 128 + k];
  s[j] = acc;
}

// ---------------------------------------------------------------- fused MLP
__global__ void __launch_bounds__(32)
fused_kernel(const float* __restrict__ x, const float* __restrict__ agg,
             const float* __restrict__ n0W, const float* __restrict__ n0b,
             const float* __restrict__ n0noise, const float* __restrict__ n0ns,
             const float* __restrict__ c1W, const float* __restrict__ c1b,
             const float* __restrict__ c2W, const float* __restrict__ c2b,
             const float* __restrict__ f0W, const float* __restrict__ f0b,
             const float* __restrict__ f0noise, const float* __restrict__ f0ns,
             const float* __restrict__ f1W, const float* __restrict__ f1b,
             const float* __restrict__ f1noise, const float* __restrict__ f1ns,
             const float* __restrict__ styleN0, const float* __restrict__ styleF0,
             const float* __restrict__ styleF1, float* __restrict__ out) {
  __shared__ float L0[16 * LDS_STRIDE];
  __shared__ float L1[16 * LDS_STRIDE];
  __shared__ float L2[16 * LDS_STRIDE];

  const int lane = threadIdx.x;           // exactly one wave32
  const int m0   = blockIdx.x * 16;
  const int col  = lane & 15;
  const int kg   = lane >> 4;

  // stage x tile (16 x 64) -> L0
  for (int i = lane; i < 16 * 16; i += 32) {
    int r = i >> 4, c4 = (i & 15) << 2;
    *(float4*)&L0[r * LDS_STRIDE + c4] = *(const float4*)&x[(size_t)(m0 + r) * IN_C + c4];
  }

  // S1: xs = style_layer(x) -> L1
  {
    const float ns = n0ns[0];
    for (int t = 0; t < 8; ++t) {
      const int n0 = t * 16;
      v8f acc = gemm_tile(L0, n0W, IN_C, n0, lane);
      const float add = n0b[n0 + col] + n0noise[n0 + col] * ns;
#pragma unroll
      for (int i = 0; i < 8; ++i)
        L1[(i + kg * 8) * LDS_STRIDE + n0 + col] = acc[i] + add;
    }
    norm_act(L1, HID, styleN0, lane);
  }

  // stage agg tile (16 x 128) -> L0
  for (int i = lane; i < 16 * 32; i += 32) {
    int r = i >> 5, c4 = (i & 31) << 2;
    *(float4*)&L0[r * LDS_STRIDE + c4] = *(const float4*)&agg[(size_t)(m0 + r) * HID + c4];
  }

  // S2: out2 = relu(agg + agg@cat1^T + b1 + xs + xs@cat2^T + b2) -> L2
  for (int t = 0; t < 8; ++t) {
    const int n0 = t * 16;
    v8f a1 = gemm_tile(L0, c1W, HID, n0, lane);
    v8f a2 = gemm_tile(L1, c2W, HID, n0, lane);
    const float add = c1b[n0 + col] + c2b[n0 + col];
#pragma unroll
    for (int i = 0; i < 8; ++i) {
      const int off = (i + kg * 8) * LDS_STRIDE + n0 + col;
      float v = a1[i] + a2[i] + add + L0[off] + L1[off];
      L2[off] = fmaxf(v, 0.f);
    }
  }

  // S3: fin0 style layer: L2 @ fin0_W^T -> L0, norm
  {
    const float ns = f0ns[0];
    for (int t = 0; t < 8; ++t) {
      const int n0 = t * 16;
      v8f acc = gemm_tile(L2, f0W, HID, n0, lane);
      const float add = f0b[n0 + col] + f0noise[n0 + col] * ns;
#pragma unroll
      for (int i = 0; i < 8; ++i)
        L0[(i + kg * 8) * LDS_STRIDE + n0 + col] = acc[i] + add;
    }
    norm_act(L0, HID, styleF0, lane);
  }

  // S4: fin1 style layer (128 -> 64), norm over 64 channels, store out
  {
    const float ns = f1ns[0];
    for (int t = 0; t < 4; ++t) {
      const int n0 = t * 16;
      v8f acc = gemm_tile(L0, f1W, HID, n0, lane);
      const float add = f1b[n0 + col] + f1noise[n0 + col] * ns;
#pragma unroll
      for (int i = 0; i < 8; ++i)
        L1[(i + kg * 8) * LDS_STRIDE + n0 + col] = acc[i] + add;
    }
    norm_act(L1, OUT_C, styleF1, lane);
    for (int i = lane; i < 16 * 16; i += 32) {
      int r = i >> 4, c4 = (i & 15) << 2;
      *(float4*)&out[(size_t)(m0 + r) * OUT_C + c4] = *(const float4*)&L1[r * LDS_STRIDE + c4];
    }
  }
}

// ---------------------------------------------------------------- launch
extern "C" void kernel_launch(void* const* d_in, const int* in_sizes, int n_in,
                              void* d_out, int out_size, void* d_ws, size_t ws_size,
                              hipStream_t stream) {
  const float* x       = (const float*)d_in[0];
  const float* w       = (const float*)d_in[1];
  const int*   edge    = (const int*)  d_in[2];
  const float* Wa      = (const float*)d_in[3];
  const float* ba      = (const float*)d_in[4];
  const float* n0W     = (const float*)d_in[5];
  const float* n0b     = (const float*)d_in[6];
  const float* n0aW    = (const float*)d_in[7];
  const float* n0ab    = (const float*)d_in[8];
  const float* n0ns    = (const float*)d_in[9];
  const float* n0noise = (const float*)d_in[10];
  const float* c1W     = (const float*)d_in[11];
  const float* c1b     = (const float*)d_in[12];
  const float* c2W     = (const float*)d_in[13];
  const float* c2b     = (const float*)d_in[14];
  const float* f0W     = (const float*)d_in[15];
  const float* f0b     = (const float*)d_in[16];
  const float* f0aW    = (const float*)d_in[17];
  const float* f0ab    = (const float*)d_in[18];
  const float* f0ns    = (const float*)d_in[19];
  const float* f0noise = (const float*)d_in[20];
  const float* f1W     = (const float*)d_in[21];
  const float* f1b     = (const float*)d_in[22];
  const float* f1aW    = (const float*)d_in[23];
  const float* f1ab    = (const float*)d_in[24];
  const float* f1ns    = (const float*)d_in[25];
  const float* f1noise = (const float*)d_in[26];

  // workspace carve-up (all f32/i32 sized)
  float* agg        = (float*)d_ws;                       // N*HID
  float* styleN0    = agg + (size_t)N_NODES * HID;        // 256
  float* styleF0    = styleN0 + 2 * HID;                  // 256
  float* styleF1    = styleF0 + 2 * HID;                  // 128
  int*   row_start  = (int*)(styleF1 + 2 * OUT_C);        // N+1
  int*   cursor     = row_start + (N_NODES + 1);          // N (deg, then cursor)
  int*   src_sorted = cursor + N_NODES;                   // E

  // CSR build: 2M int atomics total instead of 128M f32 atomics
  zero_int<<<N_NODES / 256, 256, 0, stream>>>(cursor);
  count_deg<<<N_EDGES / 256, 256, 0, stream>>>(edge, cursor);
  scan_deg<<<1, 1024, 0, stream>>>(cursor, row_start);
  fill_csr<<<N_EDGES / 256, 256, 0, stream>>>(edge, cursor, src_sorted);
  gather_sum<<<N_NODES / 8, 256, 0, stream>>>(row_start, src_sorted, Wa, ba, agg);

  style_kernel<<<1, 640, 0, stream>>>(w, n0aW, n0ab, f0aW, f0ab, f1aW, f1ab,
                                      styleN0, styleF0, styleF1);
  fused_kernel<<<N_NODES / 16, 32, 0, stream>>>(
      x, agg, n0W, n0b, n0noise, n0ns, c1W, c1b, c2W, c2b,
      f0W, f0b, f0noise, f0ns, f1W, f1b, f1noise, f1ns,
      styleN0, styleF0, styleF1, (float*)d_out);
}